// Model_73967926772512
// MI455X (gfx1250) — compile-verified
//
#include <hip/hip_runtime.h>
#include <hip/hip_bf16.h>

typedef float v2f __attribute__((ext_vector_type(2)));
typedef float v8f __attribute__((ext_vector_type(8)));

#define NUM_GRAPHS 256
#define HDIM 16

// ---------------------------------------------------------------------------
// Generic fill
// ---------------------------------------------------------------------------
__global__ void fill_kernel(float* __restrict__ p, float v, long long n) {
    long long i = (long long)blockIdx.x * blockDim.x + threadIdx.x;
    if (i < n) p[i] = v;
}

// ---------------------------------------------------------------------------
// Degree accumulation (deg pre-initialized to 1.0 for self-loops)
// ---------------------------------------------------------------------------
__global__ void deg_kernel(const int* __restrict__ dstE, float* __restrict__ deg,
                           long long E) {
    long long i = (long long)blockIdx.x * blockDim.x + threadIdx.x;
    if (i < E) atomicAdd(&deg[dstE[i]], 1.0f);
}

__global__ void rsqrt_kernel(float* __restrict__ deg, long long n) {
    long long i = (long long)blockIdx.x * blockDim.x + threadIdx.x;
    if (i < n) deg[i] = rsqrtf(deg[i]);   // in place: deg -> dinv
}

// ---------------------------------------------------------------------------
// C[N x 16] = A[N x K] * B[K x 16] via V_WMMA_F32_16X16X4_F32, K templated so
// the whole K sweep fully unrolls (32 WMMAs for K=128, 4 for K=16).
//
//   A lane layout (16x4):  lane l holds A[m=l&15][2*(l>>4) + {0,1}]
//   B lane layout (4x16):  lane l holds B[2*(l>>4) + {0,1}][n=l&15]
//   C/D layout:            acc[r] = C[r + 8*(l>>4)][l&15]
//
// Tail handling: A row m only feeds C row m, and C rows >= N are discarded at
// the store, so out-of-range lanes just CLAMP their load address (no EXEC
// divergence, no masking) and the garbage row is never written.
// ---------------------------------------------------------------------------
template <int K>
__global__ void gemm_wmma_kernel(const float* __restrict__ A,
                                 const float* __restrict__ B,
                                 float* __restrict__ C,
                                 long long N) {
    const int lane = threadIdx.x & 31;
    const int wave = threadIdx.x >> 5;
    const int wavesPerBlock = blockDim.x >> 5;
    const long long tile = (long long)blockIdx.x * wavesPerBlock + wave;
    const long long row0 = tile * 16;
    if (row0 >= N) return;                    // wave-uniform: EXEC all-1s inside

    const int n  = lane & 15;                 // B/C column; A row within tile
    const int kg = lane >> 4;                 // K-pair selector (0 or 1)
    long long arow = row0 + n;
    if (arow >= N) arow = N - 1;              // clamp only; row never stored

    const float* Ap = A + arow * K + 2 * kg;  // this lane's K-pair stream
    const float* Bp = B + (2 * kg) * HDIM + n;

    v8f acc = {};
#pragma unroll
    for (int k0 = 0; k0 < K; k0 += 4) {
        v2f a = *(const v2f*)(Ap + k0);       // global_load_b64 (8B aligned)
        v2f b;
        b.x = Bp[k0 * HDIM];
        b.y = Bp[k0 * HDIM + HDIM];
        // (neg_a, A, neg_b, B, c_mod, C, reuse_a, reuse_b)
        acc = __builtin_amdgcn_wmma_f32_16x16x4_f32(
            false, a, false, b, (short)0, acc, false, false);
    }

#pragma unroll
    for (int r = 0; r < 8; ++r) {
        long long row = row0 + r + 8 * kg;
        if (row < N) C[row * HDIM + n] = acc[r];
    }
}

// ---------------------------------------------------------------------------
// Edge scatter: out[dst] += h[src] * dinv[src]*dinv[dst], incl. N self-loops.
// One thread per (edge, 4-feature group): one b128 gather + 4 f32 atomics;
// indices/norms loaded 4x per edge instead of 16x. Atomics resolve in L2
// (out is 6.4 MB << 192 MB).
// ---------------------------------------------------------------------------
__global__ void scatter_kernel(const int* __restrict__ srcE,
                               const int* __restrict__ dstE,
                               const float* __restrict__ h,
                               const float* __restrict__ dinv,
                               float* __restrict__ out,
                               long long E, long long N) {
    long long t = (long long)blockIdx.x * blockDim.x + threadIdx.x;
    long long M = (E + N) * 4;
    if (t >= M) return;
    long long e = t >> 2;
    int f4 = (int)(t & 3) * 4;
    long long s, d;
    if (e < E) { s = srcE[e]; d = dstE[e]; }
    else       { s = d = e - E; }            // self-loop
    float nrm = dinv[s] * dinv[d];
    const float4 hv = *(const float4*)(h + s * HDIM + f4);
    float* o = out + d * HDIM + f4;
    atomicAdd(o + 0, hv.x * nrm);
    atomicAdd(o + 1, hv.y * nrm);
    atomicAdd(o + 2, hv.z * nrm);
    atomicAdd(o + 3, hv.w * nrm);
}

// ---------------------------------------------------------------------------
// h = relu(h + bias)  (layer-1 epilogue)
// ---------------------------------------------------------------------------
__global__ void bias_relu_kernel(float* __restrict__ h,
                                 const float* __restrict__ bias,
                                 long long n) {
    long long i = (long long)blockIdx.x * blockDim.x + threadIdx.x;
    if (i < n) {
        float v = h[i] + bias[i & 15];
        h[i] = fmaxf(v, 0.0f);
    }
}

// ---------------------------------------------------------------------------
// Pool: sums[batch[n]] += h[n]; counts[batch[n]] += 1
// One thread per (node, 4-feature group).
// ---------------------------------------------------------------------------
__global__ void pool_kernel(const float* __restrict__ h,
                            const int* __restrict__ batch,
                            float* __restrict__ sums,
                            float* __restrict__ counts,
                            long long N) {
    long long t = (long long)blockIdx.x * blockDim.x + threadIdx.x;
    if (t >= N * 4) return;
    long long nid = t >> 2;
    int f4 = (int)(t & 3) * 4;
    int g = batch[nid];
    const float4 hv = *(const float4*)(h + nid * HDIM + f4);
    float* o = sums + g * HDIM + f4;
    atomicAdd(o + 0, hv.x);
    atomicAdd(o + 1, hv.y);
    atomicAdd(o + 2, hv.z);
    atomicAdd(o + 3, hv.w);
    if (f4 == 0) atomicAdd(&counts[g], 1.0f);
}

// ---------------------------------------------------------------------------
// out = sigmoid(sums/max(counts,1) + b2)   (layer-2 bias folded into mean)
// ---------------------------------------------------------------------------
__global__ void final_kernel(const float* __restrict__ sums,
                             const float* __restrict__ counts,
                             const float* __restrict__ b2,
                             float* __restrict__ out) {
    int t = blockIdx.x * blockDim.x + threadIdx.x;
    if (t >= NUM_GRAPHS * HDIM) return;
    int g = t >> 4;
    int f = t & 15;
    float c = fmaxf(counts[g], 1.0f);
    float v = sums[t] / c + b2[f];
    out[t] = 1.0f / (1.0f + expf(-v));
}

// ---------------------------------------------------------------------------
static inline unsigned cdiv_u(long long a, long long b) {
    return (unsigned)((a + b - 1) / b);
}

extern "C" void kernel_launch(void* const* d_in, const int* in_sizes, int n_in,
                              void* d_out, int out_size, void* d_ws, size_t ws_size,
                              hipStream_t stream) {
    const float* x     = (const float*)d_in[0];   // [N,128]
    const int*   ei    = (const int*)d_in[1];     // [2,E]
    const int*   batch = (const int*)d_in[2];     // [N]
    const float* W1    = (const float*)d_in[3];   // [128,16]
    const float* b1    = (const float*)d_in[4];   // [16]
    const float* W2    = (const float*)d_in[5];   // [16,16]
    const float* b2    = (const float*)d_in[6];   // [16]

    const long long N = in_sizes[0] / 128;
    const long long E = in_sizes[1] / 2;
    const int* srcE = ei;
    const int* dstE = ei + E;

    // workspace layout (floats)
    float* dinv   = (float*)d_ws;                  // N
    float* hbuf   = dinv + N;                      // N*16 (GEMM out, both layers)
    float* agg    = hbuf + (size_t)N * HDIM;       // N*16 (scatter out, both layers)
    float* sums   = agg  + (size_t)N * HDIM;       // 256*16
    float* counts = sums + NUM_GRAPHS * HDIM;      // 256

    const int B = 256;
    const long long tiles = (N + 15) / 16;

    // 1) degrees (init 1.0 for self-loops) -> dinv
    fill_kernel<<<cdiv_u(N, B), B, 0, stream>>>(dinv, 1.0f, N);
    deg_kernel<<<cdiv_u(E, B), B, 0, stream>>>(dstE, dinv, E);
    rsqrt_kernel<<<cdiv_u(N, B), B, 0, stream>>>(dinv, N);

    // 2) layer 1: hbuf = x @ W1  (WMMA f32, K=128, fully unrolled)
    gemm_wmma_kernel<128><<<cdiv_u(tiles, B / 32), B, 0, stream>>>(x, W1, hbuf, N);
    fill_kernel<<<cdiv_u(N * HDIM, B), B, 0, stream>>>(agg, 0.0f, N * HDIM);
    scatter_kernel<<<cdiv_u((E + N) * 4, B), B, 0, stream>>>(srcE, dstE, hbuf,
                                                             dinv, agg, E, N);
    bias_relu_kernel<<<cdiv_u(N * HDIM, B), B, 0, stream>>>(agg, b1, N * HDIM);

    // 3) layer 2: hbuf = agg @ W2  (WMMA f32, K=16, fully unrolled)
    gemm_wmma_kernel<16><<<cdiv_u(tiles, B / 32), B, 0, stream>>>(agg, W2, hbuf, N);
    fill_kernel<<<cdiv_u(N * HDIM, B), B, 0, stream>>>(agg, 0.0f, N * HDIM);
    scatter_kernel<<<cdiv_u((E + N) * 4, B), B, 0, stream>>>(srcE, dstE, hbuf,
                                                             dinv, agg, E, N);

    // 4) mean-pool per graph + sigmoid (b2 folded into mean)
    fill_kernel<<<cdiv_u(NUM_GRAPHS * (HDIM + 1), B), B, 0, stream>>>(
        sums, 0.0f, NUM_GRAPHS * (HDIM + 1));   // zeros sums and counts (contiguous)
    pool_kernel<<<cdiv_u(N * 4, B), B, 0, stream>>>(agg, batch, sums, counts, N);
    final_kernel<<<cdiv_u(NUM_GRAPHS * HDIM, B), B, 0, stream>>>(sums, counts, b2,
                                                                 (float*)d_out);
}